// SpatialAttention_8856222564529
// MI455X (gfx1250) — compile-verified
//
#include <hip/hip_runtime.h>
#include <hip/hip_bf16.h>

// ---------------------------------------------------------------------------
// SpatialAttention SNN pipeline for gfx1250 (MI455X), wave32.
//   K1: channel-max pool  (512MB read, coalesced, memory-bound)
//   K2: LIF0 + exact top-k(50%) threshold via 4-round radix select in LDS
//   K3: 7x7 conv as banded GEMM using V_WMMA_F32_16X16X4_F32
//       - B fragments pre-materialized per-lane in LDS (no predicated loads)
//       - column apron in the spike tile (no bounds checks in inner loop)
//   K4: LIF1 (beta=0.25, vth=1.0, soft reset), membranes pre-reset
// No workspace needed: d_out (2 x 4,194,304 floats) is used as staging.
// ---------------------------------------------------------------------------

typedef float v2f __attribute__((ext_vector_type(2)));
typedef float v8f __attribute__((ext_vector_type(8)));

#define T_STEPS 16
#define B_SZ    16
#define HW      16384     // 128*128
#define BHW     262144    // 16*128*128
#define TBHW    4194304   // 16*16*128*128
#define K_RANK  8191      // k-1 for k = 16384*50/100 = 8192
#define BETA    0.25f

// ---------------- Kernel 1: max over channel dim (C=32) --------------------
__global__ __launch_bounds__(256) void pool_max_c(const float* __restrict__ x,
                                                  float* __restrict__ pooled) {
  int o = blockIdx.x * blockDim.x + threadIdx.x;   // over T*B*HW
  int pix = o & (HW - 1);
  int tb  = o >> 14;
  const float* p = x + (size_t)tb * 32 * HW + pix;
  float m = p[0];
#pragma unroll
  for (int c = 1; c < 32; ++c) m = fmaxf(m, p[c * HW]);
  pooled[o] = m;
}

// ---------------- Kernel 2: LIF0 with top-50% spiking ----------------------
// One workgroup per batch sample b (16 blocks x 1024 threads).
// mem kept in registers (16 elems/thread). Exact k-th largest via MSB-first
// 8-bit radix select on order-preserving uint keys; spike = mem >= thr.
__device__ __forceinline__ unsigned f2ord(float f) {
  unsigned u = __float_as_uint(f);
  return (u & 0x80000000u) ? ~u : (u | 0x80000000u);
}

__global__ __launch_bounds__(1024) void lif0_topk(const float* __restrict__ pooled,
                                                  float* __restrict__ spikes) {
  __shared__ unsigned hist[256];
  __shared__ unsigned sel_digit;
  __shared__ unsigned sel_rem;

  const int b   = blockIdx.x;
  const int tid = threadIdx.x;

  float mem[16];
#pragma unroll
  for (int j = 0; j < 16; ++j) mem[j] = 0.0f;

  for (int t = 0; t < T_STEPS; ++t) {
    const float* xt = pooled + (size_t)(t * B_SZ + b) * HW;
    float*       st = spikes + (size_t)(t * B_SZ + b) * HW;

    unsigned key[16];
#pragma unroll
    for (int j = 0; j < 16; ++j) {
      int i = tid + j * 1024;
      float m = BETA * mem[j] + xt[i];
      mem[j]  = m;
      key[j]  = f2ord(m);
    }

    unsigned mask = 0u, pref = 0u, rem = K_RANK;
    for (int d = 0; d < 4; ++d) {
      const int shift = 24 - 8 * d;
      if (tid < 256) hist[tid] = 0u;
      __syncthreads();
#pragma unroll
      for (int j = 0; j < 16; ++j) {
        if ((key[j] & mask) == pref)
          atomicAdd(&hist[(key[j] >> shift) & 255u], 1u);
      }
      __syncthreads();
      if (tid == 0) {
        unsigned r = rem;
        int bin = 255;
        for (; bin > 0; --bin) {
          unsigned h = hist[bin];
          if (r < h) break;
          r -= h;
        }
        sel_digit = (unsigned)bin;
        sel_rem   = r;
      }
      __syncthreads();
      pref |= sel_digit << shift;
      rem   = sel_rem;
      mask  = (mask >> 8) | 0xFF000000u;
    }

    const unsigned thr_key = pref;   // exact key of k-th largest value
#pragma unroll
    for (int j = 0; j < 16; ++j) {
      int i = tid + j * 1024;
      bool sp = (key[j] >= thr_key); // == (mem >= thr), mapping is monotonic
      st[i] = sp ? 1.0f : 0.0f;
      if (sp) mem[j] = 0.0f;         // hard reset
    }
    __syncthreads();
  }
}

// ---------------- Kernel 3: 7x7 conv via WMMA f32 16x16x4 ------------------
// Block = 256 threads (8 waves); each block does one 32-row band of one
// image (grid = 256 imgs * 4 bands). Band + 3-row halo + 3-col apron staged
// in LDS, row stride 138 (conflict-free 16-lane strided row reads).
// Per 16x16 tile: O = sum_ky A(16x24) * B(24x16), B[j][x] = w[ky][j-x].
// The 42 B fragments (per-lane v2f) depend only on (lane, ky, kc): they are
// materialized once into LDS so the inner loop is pure ds_load + wmma.
#define LSTR   138
#define NCHUNK 42          // 7 ky * 6 kc

__global__ __launch_bounds__(256) void conv7x7_wmma(const float* __restrict__ spikes,
                                                    const float* __restrict__ w,
                                                    float* __restrict__ out) {
  __shared__ float simg[38 * LSTR];        // rows -3..34 of band, cols -3..134
  __shared__ float bfrag[NCHUNK * 64];     // [chunk][lane][2], 8B-aligned pairs

  const int blk  = blockIdx.x;
  const int img  = blk >> 2;
  const int band = blk & 3;                // 32 output rows per band
  const int tid  = threadIdx.x;

  // Phase a: zero tile (apron included)
  for (int idx = tid; idx < 38 * LSTR; idx += 256) simg[idx] = 0.0f;

  // Phase a': materialize per-lane B fragments. entry e -> chunk, lane, comp
  for (int e = tid; e < NCHUNK * 64; e += 256) {
    int chunk = e >> 6;                    // 0..41
    int lp    = e & 63;
    int ln    = lp >> 1;                   // lane 0..31
    int comp  = lp & 1;
    int ky    = chunk / 6;
    int kc    = chunk - ky * 6;
    int halfb = ln >> 4;
    int lob   = ln & 15;
    int j     = 4 * kc + 2 * halfb + comp; // K index within 0..23
    int kx    = j - lob;                   // B[j][x=lo] = w[ky][j-lo]
    bfrag[e]  = (kx >= 0 && kx <= 6) ? w[ky * 7 + kx] : 0.0f;
  }
  __syncthreads();

  // Phase b: fill interior (cols 0..127 at index c+3; rows haloed by 3)
  const float* sp = spikes + (size_t)img * HW;
  const int row_base = band * 32 - 3;
  for (int idx = tid; idx < 38 * 128; idx += 256) {
    int rr = idx >> 7;
    int c  = idx & 127;
    int rg = row_base + rr;
    if (rg >= 0 && rg < 128) simg[rr * LSTR + c + 3] = sp[rg * 128 + c];
  }
  __syncthreads();

  const int wave = tid >> 5;
  const int lane = tid & 31;
  const int half = lane >> 4;              // 0: lanes 0-15, 1: lanes 16-31
  const int lo   = lane & 15;

  for (int ti = 0; ti < 2; ++ti) {         // 16 tiles / 8 waves
    const int tile   = wave * 2 + ti;      // 0..15 (2 tile-rows x 8 tile-cols)
    const int ylocal = (tile >> 3) << 4;
    const int x0     = (tile & 7) << 4;

    v8f acc = {};
    for (int ky = 0; ky < 7; ++ky) {
      // A row for this lane: M = lo -> padded row ylocal+lo+ky, col j+x0
      const float* rowp = simg + (ylocal + lo + ky) * LSTR + x0;
#pragma unroll
      for (int kc = 0; kc < 6; ++kc) {
        const int j0 = 4 * kc + 2 * half;  // K index of a.x
        v2f a, bm;
        a.x = rowp[j0];                    // apron-padded: no bounds checks
        a.y = rowp[j0 + 1];
        bm  = *(const v2f*)(bfrag + ((ky * 6 + kc) << 6) + (lane << 1));
        acc = __builtin_amdgcn_wmma_f32_16x16x4_f32(
            false, a, false, bm, (short)0, acc, false, false);
      }
    }

    float* op = out + (size_t)img * HW + (band * 32 + ylocal) * 128 + x0;
#pragma unroll
    for (int i = 0; i < 8; ++i)            // D VGPR i -> row i + 8*half
      op[(i + 8 * half) * 128 + lo] = acc[i];
  }
}

// ---------------- Kernel 4: LIF1 (vth=1, soft reset) -----------------------
// Reads conv values from region A and overwrites in place (read-before-write
// per thread/offset); membranes (pre-reset) go to region B.
__global__ __launch_bounds__(256) void lif1_k(float* __restrict__ data) {
  const int idx = blockIdx.x * blockDim.x + threadIdx.x;  // < BHW
  float* spike_out = data;
  float* mem_out   = data + TBHW;
  float mem = 0.0f;
  for (int t = 0; t < T_STEPS; ++t) {
    size_t off = (size_t)t * BHW + idx;
    float v = spike_out[off];          // conv value staged here
    mem = BETA * mem + v;
    float sp = (mem > 1.0f) ? 1.0f : 0.0f;
    spike_out[off] = sp;
    mem_out[off]   = mem;              // pre-reset membrane
    mem -= sp;                         // soft reset (vth = 1.0)
  }
}

// ---------------------------------------------------------------------------
extern "C" void kernel_launch(void* const* d_in, const int* in_sizes, int n_in,
                              void* d_out, int out_size, void* d_ws, size_t ws_size,
                              hipStream_t stream) {
  (void)in_sizes; (void)n_in; (void)out_size; (void)d_ws; (void)ws_size;
  const float* x      = (const float*)d_in[0];   // (16,16,32,128,128) f32
  const float* conv_w = (const float*)d_in[1];   // (1,1,7,7) f32
  float* out = (float*)d_out;                    // 2 * 4194304 f32

  float* regionA = out;          // pooled -> conv -> final spikes
  float* regionB = out + TBHW;   // lif0 spikes -> final membranes

  pool_max_c  <<<TBHW / 256, 256, 0, stream>>>(x, regionA);
  lif0_topk   <<<B_SZ, 1024, 0, stream>>>(regionA, regionB);
  conv7x7_wmma<<<T_STEPS * B_SZ * 4, 256, 0, stream>>>(regionB, conv_w, regionA);
  lif1_k      <<<BHW / 256, 256, 0, stream>>>(out);
}